// pdfam_gmm_module_69303592288979
// MI455X (gfx1250) — compile-verified
//
#include <hip/hip_runtime.h>
#include <cstdint>
#include <cstddef>

// ---------------------------------------------------------------------------
// GMM (K=2, 3 EM iters) per (b,c) channel, N = 16384 samples/channel.
// One 256-thread workgroup per channel; whole channel resident in 64KB LDS
// (MI455X: 320KB LDS/WGP). Global memory read once (async global->LDS DMA),
// written once. All reductions are wave32 shuffles + tiny LDS scratch.
// ---------------------------------------------------------------------------

#define F_LOG2PI 1.8378770664093453f
#define F_EPS    1e-6f
#define F_TWO_PI 6.283185307179586f

constexpr int kC      = 128;          // channels
constexpr int kN      = 16384;        // samples per channel (h*w)
constexpr int kBlk    = 256;          // threads per block (8 waves of 32)
constexpr int kWaves  = kBlk / 32;    // 8
constexpr int kVec    = kN / (kBlk * 4); // 16 float4 chunks per thread
constexpr int kIters  = 3;            // NUM_T
// NUM_K = 2 hardcoded throughout.

__device__ __forceinline__ float softplus_f(float z) {
    return __logf(1.0f + __expf(z));
}

// Block-wide reduce of (sum, sum, sum) across 256 threads (wave32).
__device__ __forceinline__ void blk_sum3(float& a, float& b, float& c,
                                         float* scr, int tid) {
#pragma unroll
    for (int m = 16; m > 0; m >>= 1) {
        a += __shfl_xor(a, m, 32);
        b += __shfl_xor(b, m, 32);
        c += __shfl_xor(c, m, 32);
    }
    const int wv = tid >> 5, ln = tid & 31;
    if (ln == 0) { scr[wv * 3 + 0] = a; scr[wv * 3 + 1] = b; scr[wv * 3 + 2] = c; }
    __syncthreads();
    if (tid < 32) {
        const bool ok = ln < kWaves;
        float x = ok ? scr[ln * 3 + 0] : 0.0f;
        float y = ok ? scr[ln * 3 + 1] : 0.0f;
        float z = ok ? scr[ln * 3 + 2] : 0.0f;
#pragma unroll
        for (int m = 16; m > 0; m >>= 1) {
            x += __shfl_xor(x, m, 32);
            y += __shfl_xor(y, m, 32);
            z += __shfl_xor(z, m, 32);
        }
        if (ln == 0) { scr[0] = x; scr[1] = y; scr[2] = z; }
    }
    __syncthreads();
    a = scr[0]; b = scr[1]; c = scr[2];
    __syncthreads();   // scratch reusable after return
}

// Block-wide reduce of (sum, sumsq, min, max).
__device__ __forceinline__ void blk_stats4(float& s, float& q, float& mn, float& mx,
                                           float* scr, int tid) {
#pragma unroll
    for (int m = 16; m > 0; m >>= 1) {
        s += __shfl_xor(s, m, 32);
        q += __shfl_xor(q, m, 32);
        mn = fminf(mn, __shfl_xor(mn, m, 32));
        mx = fmaxf(mx, __shfl_xor(mx, m, 32));
    }
    const int wv = tid >> 5, ln = tid & 31;
    if (ln == 0) {
        scr[wv * 4 + 0] = s;  scr[wv * 4 + 1] = q;
        scr[wv * 4 + 2] = mn; scr[wv * 4 + 3] = mx;
    }
    __syncthreads();
    if (tid < 32) {
        const bool ok = ln < kWaves;
        float a = ok ? scr[ln * 4 + 0] : 0.0f;
        float b = ok ? scr[ln * 4 + 1] : 0.0f;
        float c = ok ? scr[ln * 4 + 2] : 3.4e38f;
        float d = ok ? scr[ln * 4 + 3] : -3.4e38f;
#pragma unroll
        for (int m = 16; m > 0; m >>= 1) {
            a += __shfl_xor(a, m, 32);
            b += __shfl_xor(b, m, 32);
            c = fminf(c, __shfl_xor(c, m, 32));
            d = fmaxf(d, __shfl_xor(d, m, 32));
        }
        if (ln == 0) { scr[0] = a; scr[1] = b; scr[2] = c; scr[3] = d; }
    }
    __syncthreads();
    s = scr[0]; q = scr[1]; mn = scr[2]; mx = scr[3];
    __syncthreads();
}

__global__ __launch_bounds__(kBlk)
void pdfam_gmm_kernel(const float* __restrict__ x,
                      const float* __restrict__ learn_eps,
                      const float* __restrict__ learn_coef,
                      const float* __restrict__ learn_width,
                      float* __restrict__ out) {
    extern __shared__ float smem[];        // kN floats = 64KB, one channel
    __shared__ float scr[32];              // reduction scratch

    const int ch  = blockIdx.x;            // 0..1023  (= b*128 + c)
    const int cc  = ch & (kC - 1);         // channel index for learn_* params
    const int tid = threadIdx.x;

    const float* __restrict__ xc = x   + (size_t)ch * kN;
    float*       __restrict__ oc = out + (size_t)ch * kN;

    // ---- Phase 0: async bulk copy global -> LDS (CDNA5 ASYNCcnt path) ----
    // Each lane issues kVec b128 DMA ops; LDS dest addr = low 32 bits of the
    // generic shared pointer.
#pragma unroll
    for (int i = 0; i < kVec; ++i) {
        const int e = i * (kBlk * 4) + tid * 4;
        const float* gp = xc + e;
        unsigned ldsoff = (unsigned)(uintptr_t)(smem + e);
        asm volatile("global_load_async_to_lds_b128 %0, %1, off"
                     :: "v"(ldsoff), "v"(gp) : "memory");
    }
    asm volatile("s_wait_asynccnt 0x0" ::: "memory");
    __syncthreads();

    // ---- Phase 1: init stats (sum, sumsq, min, max) over the channel ----
    float s_all = 0.0f, q_all = 0.0f, mn = 3.4e38f, mx = -3.4e38f;
#pragma unroll 4
    for (int i = 0; i < kVec; ++i) {
        const float4 v = *reinterpret_cast<const float4*>(smem + i * (kBlk * 4) + tid * 4);
        s_all += v.x + v.y + v.z + v.w;
        q_all = fmaf(v.x, v.x, fmaf(v.y, v.y, fmaf(v.z, v.z, fmaf(v.w, v.w, q_all))));
        mn = fminf(mn, fminf(fminf(v.x, v.y), fminf(v.z, v.w)));
        mx = fmaxf(mx, fmaxf(fmaxf(v.x, v.y), fmaxf(v.z, v.w)));
    }
    blk_stats4(s_all, q_all, mn, mx, scr, tid);

    const float invN  = 1.0f / (float)kN;
    const float mean  = s_all * invN;
    float var_init    = fmaxf(q_all * invN - mean * mean, 0.0f) + F_EPS;

    // K=2 init: mu evenly spaced in [min,max]; shared var; uniform pi.
    float mu0 = mn + (mx - mn) * (1.0f / 3.0f);
    float mu1 = mn + (mx - mn) * (2.0f / 3.0f);
    float v0 = var_init, v1 = var_init;
    float pi0 = 0.5f, pi1 = 0.5f;

    // ---- Phase 2: EM iterations (data stays in LDS) ----
    for (int it = 0; it < kIters; ++it) {
        // delta = logp0 - logp1 = A0*(x-mu0)^2 - A1*(x-mu1)^2 + cst
        const float A0  = -0.5f / v0;
        const float A1  = -0.5f / v1;
        const float cst = -0.5f * __logf(v0) + __logf(pi0)
                        +  0.5f * __logf(v1) - __logf(pi1);

        float n0 = 0.0f, s0 = 0.0f, q0 = 0.0f;   // Σr0, Σr0·x, Σr0·x²
#pragma unroll 4
        for (int i = 0; i < kVec; ++i) {
            const float4 v = *reinterpret_cast<const float4*>(smem + i * (kBlk * 4) + tid * 4);
            const float xv[4] = { v.x, v.y, v.z, v.w };
#pragma unroll
            for (int j = 0; j < 4; ++j) {
                const float xf = xv[j];
                const float d0 = xf - mu0;
                const float d1 = xf - mu1;
                const float delta = fmaf(A0 * d0, d0, cst) - A1 * d1 * d1;
                const float r0 = 1.0f / (1.0f + __expf(-delta));  // softmax_0 of K=2
                n0 += r0;
                s0 = fmaf(r0, xf, s0);
                q0 = fmaf(r0 * xf, xf, q0);
            }
        }
        blk_sum3(n0, s0, q0, scr, tid);

        // Component 1 sums via complements (r0 + r1 == 1).
        const float n1 = (float)kN - n0;
        const float s1 = s_all - s0;
        const float q1 = q_all - q0;

        const float Nk0 = n0 + F_EPS, Nk1 = n1 + F_EPS;
        pi0 = Nk0 * invN;             pi1 = Nk1 * invN;
        mu0 = s0 / Nk0;               mu1 = s1 / Nk1;
        // Σr(x-mu)² = Q - 2·mu·S + mu²·Σr
        v0 = fmaxf((q0 - 2.0f * mu0 * s0 + mu0 * mu0 * n0) / Nk0, 0.0f) + F_EPS;
        v1 = fmaxf((q1 - 2.0f * mu1 * s1 + mu1 * mu1 * n1) / Nk1, 0.0f) + F_EPS;
    }

    // ---- Phase 3: evaluate mixture pdf and modulate x ----
    const float widthC = softplus_f(learn_width[cc]);
    const float w2     = widthC * widthC;
    const float ve0    = v0 * w2 + F_EPS;
    const float ve1    = v1 * w2 + F_EPS;
    const float amp0   = pi0 * __frsqrt_rn(F_TWO_PI * ve0);
    const float amp1   = pi1 * __frsqrt_rn(F_TWO_PI * ve1);
    const float e0     = -0.5f / ve0;
    const float e1     = -0.5f / ve1;
    const float eps_l  = softplus_f(learn_eps[cc]);
    const float coef_l = softplus_f(learn_coef[cc]);

#pragma unroll 4
    for (int i = 0; i < kVec; ++i) {
        const int e = i * (kBlk * 4) + tid * 4;
        const float4 v = *reinterpret_cast<const float4*>(smem + e);
        float r[4];
        const float xv[4] = { v.x, v.y, v.z, v.w };
#pragma unroll
        for (int j = 0; j < 4; ++j) {
            const float xf  = xv[j];
            const float d0  = xf - mu0;
            const float d1  = xf - mu1;
            const float pdf = amp0 * __expf(e0 * d0 * d0)
                            + amp1 * __expf(e1 * d1 * d1);
            r[j] = xf / (fmaf(pdf, coef_l, eps_l) + 1.0f);
        }
        float4 o; o.x = r[0]; o.y = r[1]; o.z = r[2]; o.w = r[3];
        *reinterpret_cast<float4*>(oc + e) = o;
    }
}

extern "C" void kernel_launch(void* const* d_in, const int* in_sizes, int n_in,
                              void* d_out, int out_size, void* d_ws, size_t ws_size,
                              hipStream_t stream) {
    (void)in_sizes; (void)n_in; (void)d_ws; (void)ws_size;
    const float* x      = (const float*)d_in[0];
    const float* leps   = (const float*)d_in[1];
    const float* lcoef  = (const float*)d_in[2];
    const float* lwidth = (const float*)d_in[3];
    float* out          = (float*)d_out;

    const int nch = out_size / kN;               // 1024 channels
    const size_t shmem = (size_t)kN * sizeof(float); // 64KB dynamic LDS / block

    hipLaunchKernelGGL(pdfam_gmm_kernel, dim3(nch), dim3(kBlk), shmem, stream,
                       x, leps, lcoef, lwidth, out);
}